// LIF_88098369175835
// MI455X (gfx1250) — compile-verified
//
#include <hip/hip_runtime.h>
#include <cstdint>
#include <cstddef>

// LIF scan: x[B=64, C=2048, T=256] f32 -> spikes same shape.
// Memory-bound streaming scan; CDNA5 path = async global->LDS staging (ASYNCcnt),
// NT cache policy on both streams (268MB working set > 192MB L2).

#define LIF_TAU 0.25f
#define LIF_VTH 0.5f

typedef float v4f __attribute__((ext_vector_type(4)));   // native vector for NT b128 store

constexpr int T       = 256;
constexpr int TC      = 64;          // timesteps per chunk
constexpr int NCHK    = T / TC;      // 4 chunks
constexpr int ROWS    = 128;         // channels (rows) per block
constexpr int STR     = TC + 1;      // 65-float LDS row stride (odd -> conflict-free)
constexpr int NROWTOT = 64 * 2048;   // 131072 channels

__global__ __launch_bounds__(ROWS) void lif_scan_kernel(const float* __restrict__ x,
                                                        float* __restrict__ out) {
    __shared__ float sbuf[2][ROWS * STR];   // double-buffered tile, ~65 KB

    const int tid = threadIdx.x;
    const float* xg = x   + (size_t)blockIdx.x * (size_t)ROWS * T;
    float*       og = out + (size_t)blockIdx.x * (size_t)ROWS * T;

    // Cooperative coalesced tile load: 16 consecutive lanes fetch one row's
    // 256B chunk as 16x b128; each thread issues 16 async loads per chunk.
    const int seg = tid & 15;   // which 16B segment of the 256B row-chunk
    const int rb  = tid >> 4;   // base row 0..7, stepped by 8

    auto issue_chunk = [&](int chunk, int sel) {
        const int t0 = chunk * TC;
        #pragma unroll
        for (int i = 0; i < 16; ++i) {
            const int row = rb + i * 8;
            unsigned voff = (unsigned)((row * T + t0 + seg * 4) * (int)sizeof(float));
            // low 32 bits of a generic pointer to LDS == workgroup-relative LDS byte offset
            unsigned lofs = (unsigned)(uintptr_t)(&sbuf[sel][row * STR + seg * 4]);
            asm volatile("global_load_async_to_lds_b128 %0, %1, %2 th:TH_LOAD_NT"
                         :: "v"(lofs), "v"(voff), "s"(xg)
                         : "memory");
        }
    };

    issue_chunk(0, 0);

    float u = 0.0f;        // membrane potential carry
    bool  fired = false;   // previous spike carry (s_{t-1} in {0,1})

    #pragma unroll
    for (int c = 0; c < NCHK; ++c) {
        const int sel = c & 1;
        if (c + 1 < NCHK) {
            issue_chunk(c + 1, sel ^ 1);
            // 16 loads of chunk c+1 outstanding; wait for chunk c's 16 to land
            asm volatile("s_wait_asynccnt 16" ::: "memory");
        } else {
            asm volatile("s_wait_asynccnt 0" ::: "memory");
        }
        __syncthreads();   // everyone's portion of the tile is now in LDS

        // lane-per-row scan; stride-65 rows -> 32 lanes hit 32 distinct banks
        const float* rowp = &sbuf[sel][tid * STR];
        v4f ov;
        #pragma unroll
        for (int t = 0; t < TC; ++t) {
            float xt = rowp[t];
            float ug = fired ? 0.0f : u;        // reset:  u * (1 - s_prev)
            u = fmaf(LIF_TAU, ug, xt);          // leak + integrate
            fired = (u > LIF_VTH);              // fire
            ov[t & 3] = fired ? 1.0f : 0.0f;
            if ((t & 3) == 3) {                 // NT b128 store, write-once stream
                __builtin_nontemporal_store(ov,
                    (v4f*)(og + (size_t)tid * T + c * TC + (t - 3)));
            }
        }
        __syncthreads();   // tile buffer free for reuse before next issue
    }
}

extern "C" void kernel_launch(void* const* d_in, const int* in_sizes, int n_in,
                              void* d_out, int out_size, void* d_ws, size_t ws_size,
                              hipStream_t stream) {
    (void)in_sizes; (void)n_in; (void)out_size; (void)d_ws; (void)ws_size;
    const float* x = (const float*)d_in[0];
    float* out = (float*)d_out;

    dim3 grid(NROWTOT / ROWS);   // 1024 workgroups
    dim3 block(ROWS);            // 128 threads = 4 wave32
    hipLaunchKernelGGL(lif_scan_kernel, grid, block, 0, stream, x, out);
}